// MambaLM_59562606461545
// MI455X (gfx1250) — compile-verified
//
#include <hip/hip_runtime.h>
#include <math.h>

// ---- model dims ----
#define L_  24
#define B_  32
#define D_  1024
#define ED_ 2048
#define N_  16
#define R_  64
#define K_  4
#define V_  50280
#define EPS_ 1e-5f

typedef __attribute__((ext_vector_type(2))) float v2f;
typedef __attribute__((ext_vector_type(8))) float v8f;
typedef __attribute__((ext_vector_type(4))) int   v4i;

typedef __attribute__((address_space(1))) v4i* g_v4i_p;  // global int4*
typedef __attribute__((address_space(3))) v4i* l_v4i_p;  // LDS int4*

#if defined(__has_builtin)
#if __has_builtin(__builtin_amdgcn_global_load_async_to_lds_b128) && \
    __has_builtin(__builtin_amdgcn_s_wait_asynccnt)
#define HAVE_ASYNC_LDS 1
#endif
#endif
#ifndef HAVE_ASYNC_LDS
#define HAVE_ASYNC_LDS 0
#endif

__device__ __forceinline__ float silu_f(float x) {
    return x / (1.0f + __expf(-x));
}
__device__ __forceinline__ float softplus_f(float x) {
    return (x > 20.0f) ? x : log1pf(__expf(x));
}

__device__ __forceinline__ v8f wmma_f32(v2f a, v2f b, v8f c) {
    return __builtin_amdgcn_wmma_f32_16x16x4_f32(
        false, a, false, b, (short)0, c, false, false);
}

// 8 contiguous floats of one D-fragment -> two b128 stores
__device__ __forceinline__ void store_frag(float* base, int col, const v8f& c) {
    float4 lo; lo.x = c[0]; lo.y = c[1]; lo.z = c[2]; lo.w = c[3];
    float4 hi; hi.x = c[4]; hi.y = c[5]; hi.z = c[6]; hi.w = c[7];
    *(float4*)(base + col)     = lo;
    *(float4*)(base + col + 4) = hi;
}

// ---------------------------------------------------------------------------
// Staged big-GEMM: Y[b][o] = sum_k X[b][k]*W[o][k], one wave per 32 output
// rows, batch 32 = two N=16 tiles. A operand (weights, the HBM stream) is
// double-buffered through LDS with global_load_async_to_lds_b128 (contiguous
// 16B per lane, 512B per instruction), synchronized with s_wait_asynccnt,
// consumed with conflict-free ds_load_b64 (row stride 68 floats).
// Requires K % 64 == 0. Plain epilogue (used by in_proj and logits).
// ---------------------------------------------------------------------------
#define KB_ 64
#define AST_ 68   // LDS row stride in floats (272B: 16B-aligned, bank shift 4)

__global__ __launch_bounds__(32)
void gemm_wmma_staged(const float* __restrict__ X, int ldx,
                      const float* __restrict__ W,
                      float* __restrict__ Y,
                      int O, int K)
{
    const int lane = threadIdx.x;      // 0..31
    const int o0   = blockIdx.x * 32;
    const int m    = lane & 15;
    const int g    = lane >> 4;
    const int kp   = g * 2;

    __shared__ float atile[2][32 * AST_];

    // async-fetch lane mapping: lane covers (row = 2*rr + g, cols m*4..m*4+3)
    const int fcol = m * 4;

    // B operands stream straight from L2-resident activations
    const float* x0 = X + (size_t)m        * ldx + kp;
    const float* x1 = X + (size_t)(m + 16) * ldx + kp;

    v8f c00 = {}, c01 = {}, c10 = {}, c11 = {};

    const int nkb = K / KB_;

    auto issue_block = [&](int kb, int buf) {
#if HAVE_ASYNC_LDS
        #pragma unroll
        for (int rr = 0; rr < 16; ++rr) {
            int row = o0 + 2 * rr + g;
            if (row > O - 1) row = O - 1;
            const float* gsrc = W + (size_t)row * K + kb * KB_ + fcol;
            float* ldst = &atile[buf][(2 * rr + g) * AST_ + fcol];
            __builtin_amdgcn_global_load_async_to_lds_b128(
                (g_v4i_p)(void*)gsrc,
                (l_v4i_p)(void*)ldst,
                0, 0);
        }
#else
        #pragma unroll
        for (int rr = 0; rr < 16; ++rr) {
            int row = o0 + 2 * rr + g;
            if (row > O - 1) row = O - 1;
            const float4 v = *(const float4*)(W + (size_t)row * K + kb * KB_ + fcol);
            *(float4*)&atile[buf][(2 * rr + g) * AST_ + fcol] = v;
        }
#endif
    };

    auto compute_block = [&](int kb, int buf) {
        const float* a0p = &atile[buf][(size_t)m        * AST_ + kp];
        const float* a1p = &atile[buf][(size_t)(m + 16) * AST_ + kp];
        const float* bx0 = x0 + (size_t)kb * KB_;
        const float* bx1 = x1 + (size_t)kb * KB_;
        #pragma unroll
        for (int kk = 0; kk < KB_; kk += 4) {
            v2f a0 = *(const v2f*)(a0p + kk);
            v2f a1 = *(const v2f*)(a1p + kk);
            v2f b0 = *(const v2f*)(bx0 + kk);
            v2f b1 = *(const v2f*)(bx1 + kk);
            c00 = wmma_f32(a0, b0, c00);
            c01 = wmma_f32(a0, b1, c01);
            c10 = wmma_f32(a1, b0, c10);
            c11 = wmma_f32(a1, b1, c11);
        }
    };

    issue_block(0, 0);
    for (int kb = 0; kb + 1 < nkb; ++kb) {
        const int buf = kb & 1;
        issue_block(kb + 1, buf ^ 1);
#if HAVE_ASYNC_LDS
        __builtin_amdgcn_s_wait_asynccnt(16);  // block kb fully landed in LDS
#endif
        compute_block(kb, buf);
    }
#if HAVE_ASYNC_LDS
    __builtin_amdgcn_s_wait_asynccnt(0);
#endif
    compute_block(nkb - 1, (nkb - 1) & 1);

    // D layout: element r of lane L -> (col = o0 + 8*(L>>4) + r, batch = L&15)
    if (o0 + 32 <= O) {
        store_frag(Y + (size_t)m        * O, o0 + 8 * g,      c00);
        store_frag(Y + (size_t)(m + 16) * O, o0 + 8 * g,      c01);
        store_frag(Y + (size_t)m        * O, o0 + 16 + 8 * g, c10);
        store_frag(Y + (size_t)(m + 16) * O, o0 + 16 + 8 * g, c11);
    } else {
        #pragma unroll
        for (int r = 0; r < 8; ++r) {
            int oA = o0 + 8 * g + r;
            int oB = oA + 16;
            if (oA < O) {
                Y[(size_t)m        * O + oA] = c00[r];
                Y[(size_t)(m + 16) * O + oA] = c01[r];
            }
            if (oB < O) {
                Y[(size_t)m        * O + oB] = c10[r];
                Y[(size_t)(m + 16) * O + oB] = c11[r];
            }
        }
    }
}

// ---------------------------------------------------------------------------
// Direct small-GEMM (one wave per 16 rows, operands register-streamed).
// MODE: 0 = plain, 1 = softplus(Y + bias), 2 = Y + residual
// ---------------------------------------------------------------------------
template <int MODE>
__global__ __launch_bounds__(32)
void gemm_wmma_direct(const float* __restrict__ X, int ldx,
                      const float* __restrict__ W,
                      const float* __restrict__ bias,
                      const float* __restrict__ res,
                      float* __restrict__ Y,
                      int O, int K)
{
    const int lane = threadIdx.x;
    const int o0   = blockIdx.x * 16;
    const int m    = lane & 15;
    const int g    = lane >> 4;
    const int kp   = g * 2;

    int row = o0 + m;
    if (row > O - 1) row = O - 1;

    const float* wrow = W + (size_t)row * K + kp;
    const float* x0   = X + (size_t)m        * ldx + kp;
    const float* x1   = X + (size_t)(m + 16) * ldx + kp;

    v8f c0 = {};
    v8f c1 = {};
    for (int k = 0; k < K; k += 4) {
        v2f a  = *(const v2f*)(wrow + k);
        v2f b0 = *(const v2f*)(x0 + k);
        v2f b1 = *(const v2f*)(x1 + k);
        c0 = wmma_f32(a, b0, c0);
        c1 = wmma_f32(a, b1, c1);
    }

    if (o0 + 16 <= O) {
        v8f v0 = c0, v1 = c1;
        #pragma unroll
        for (int r = 0; r < 8; ++r) {
            int o = o0 + 8 * g + r;
            if (MODE == 1) {
                float bb = bias[o];
                v0[r] = softplus_f(v0[r] + bb);
                v1[r] = softplus_f(v1[r] + bb);
            }
            if (MODE == 2) {
                v0[r] += res[(size_t)m        * O + o];
                v1[r] += res[(size_t)(m + 16) * O + o];
            }
        }
        store_frag(Y + (size_t)m        * O, o0 + 8 * g, v0);
        store_frag(Y + (size_t)(m + 16) * O, o0 + 8 * g, v1);
    } else {
        #pragma unroll
        for (int r = 0; r < 8; ++r) {
            int o = o0 + 8 * g + r;
            if (o < O) {
                float v0 = c0[r], v1 = c1[r];
                if (MODE == 1) {
                    float bb = bias[o];
                    v0 = softplus_f(v0 + bb);
                    v1 = softplus_f(v1 + bb);
                }
                if (MODE == 2) {
                    v0 += res[(size_t)m        * O + o];
                    v1 += res[(size_t)(m + 16) * O + o];
                }
                Y[(size_t)m        * O + o] = v0;
                Y[(size_t)(m + 16) * O + o] = v1;
            }
        }
    }
}

// ---------------------------------------------------------------------------
__global__ __launch_bounds__(256)
void embed_kernel(const int* __restrict__ token,
                  const float* __restrict__ emb,
                  float* __restrict__ x)
{
    int i = blockIdx.x * 256 + threadIdx.x;
    if (i < B_ * D_) {
        int b = i / D_;
        int d = i - b * D_;
        x[i] = emb[(size_t)token[b] * D_ + d];
    }
}

// ---------------------------------------------------------------------------
__global__ __launch_bounds__(256)
void rmsnorm_kernel(const float* __restrict__ x,
                    const float* __restrict__ w,
                    float* __restrict__ y, int Dd)
{
    __shared__ float red[256];
    const int b = blockIdx.x;
    const int t = threadIdx.x;
    const float* xr = x + (size_t)b * Dd;

    float s = 0.0f;
    for (int d = t; d < Dd; d += 256) {
        float v = xr[d];
        s += v * v;
    }
    red[t] = s;
    __syncthreads();
    for (int off = 128; off > 0; off >>= 1) {
        if (t < off) red[t] += red[t + off];
        __syncthreads();
    }
    float scale = rsqrtf(red[0] / (float)Dd + EPS_);
    for (int d = t; d < Dd; d += 256)
        y[(size_t)b * Dd + d] = xr[d] * scale * w[d];
}

// ---------------------------------------------------------------------------
__global__ __launch_bounds__(256)
void conv_kernel(const float* __restrict__ xz,      // (B, 2*ED), xi = first half
                 const float* __restrict__ ci,      // (B, ED, 3)
                 const float* __restrict__ cw,      // (ED, 4)
                 const float* __restrict__ cb,      // (ED,)
                 float* __restrict__ xc,            // (B, ED)
                 float* __restrict__ ci_out)        // (B, ED, 3) in d_out
{
    int i = blockIdx.x * 256 + threadIdx.x;
    if (i >= B_ * ED_) return;
    int b = i / ED_;
    int e = i - b * ED_;

    float xi = xz[(size_t)b * (2 * ED_) + e];
    const float* cip = ci + ((size_t)b * ED_ + e) * (K_ - 1);
    const float* cwp = cw + (size_t)e * K_;

    float c0 = cip[0], c1 = cip[1], c2 = cip[2];
    float acc = c0 * cwp[0] + c1 * cwp[1] + c2 * cwp[2] + xi * cwp[3] + cb[e];
    xc[i] = silu_f(acc);

    float* cop = ci_out + ((size_t)b * ED_ + e) * (K_ - 1);
    cop[0] = c1;
    cop[1] = c2;
    cop[2] = xi;
}

// ---------------------------------------------------------------------------
__global__ __launch_bounds__(256)
void ssm_kernel(const float* __restrict__ delta,   // (B, ED)
                const float* __restrict__ xc,      // (B, ED)
                const float* __restrict__ xz,      // (B, 2*ED), z = second half
                const float* __restrict__ dbc,     // (B, 96): [dt|B|C]
                const float* __restrict__ alog,    // (ED, N)
                const float* __restrict__ dparam,  // (ED,)
                const float* __restrict__ hin,     // (B, ED, N)
                float* __restrict__ hout,          // (B, ED, N) in d_out
                float* __restrict__ u)             // (B, ED)
{
    int i = blockIdx.x * 256 + threadIdx.x;
    if (i >= B_ * ED_) return;
    int b = i / ED_;
    int e = i - b * ED_;

    float dt  = delta[i];
    float xcv = xc[i];
    float dd  = dparam[e];
    const float* Bm = dbc + (size_t)b * (R_ + 2 * N_) + R_;
    const float* Cm = Bm + N_;
    const float* al = alog + (size_t)e * N_;
    const float* hp = hin + ((size_t)b * ED_ + e) * N_;
    float* ho       = hout + ((size_t)b * ED_ + e) * N_;

    float y = 0.0f;
    #pragma unroll
    for (int n = 0; n < N_; ++n) {
        float A  = -__expf(al[n]);
        float dA = __expf(dt * A);
        float hn = dA * hp[n] + dt * Bm[n] * xcv;
        ho[n] = hn;
        y += hn * Cm[n];
    }
    y += dd * xcv;

    float z = xz[(size_t)b * (2 * ED_) + ED_ + e];
    u[i] = y * silu_f(z);
}

// ---------------------------------------------------------------------------
extern "C" void kernel_launch(void* const* d_in, const int* in_sizes, int n_in,
                              void* d_out_v, int out_size, void* d_ws, size_t ws_size,
                              hipStream_t stream)
{
    (void)in_sizes; (void)n_in; (void)out_size; (void)ws_size;

    const int*   token       = (const int*)  d_in[0];
    const float* hs          = (const float*)d_in[1];
    const float* conv_inputs = (const float*)d_in[2];
    const float* embedding   = (const float*)d_in[3];
    const float* norm_w      = (const float*)d_in[4];
    const float* in_proj_w   = (const float*)d_in[5];
    const float* conv_w      = (const float*)d_in[6];
    const float* conv_b      = (const float*)d_in[7];
    const float* x_proj_w    = (const float*)d_in[8];
    const float* dt_proj_w   = (const float*)d_in[9];
    const float* dt_proj_b   = (const float*)d_in[10];
    const float* A_log       = (const float*)d_in[11];
    const float* D_param     = (const float*)d_in[12];
    const float* out_proj_w  = (const float*)d_in[13];
    const float* norm_f_w    = (const float*)d_in[14];
    float* out = (float*)d_out_v;

    // workspace layout (floats)
    float* ws    = (float*)d_ws;
    float* xA    = ws; ws += B_ * D_;
    float* xB    = ws; ws += B_ * D_;
    float* xn    = ws; ws += B_ * D_;
    float* xz    = ws; ws += B_ * 2 * ED_;
    float* xc    = ws; ws += B_ * ED_;
    float* dbc   = ws; ws += B_ * (R_ + 2 * N_);
    float* delta = ws; ws += B_ * ED_;
    float* u     = ws; ws += B_ * ED_;

    const size_t HS_BASE = (size_t)B_ * V_;
    const size_t CI_BASE = HS_BASE + (size_t)L_ * B_ * ED_ * N_;

    embed_kernel<<<(B_ * D_ + 255) / 256, 256, 0, stream>>>(token, embedding, xA);

    float* xcur  = xA;
    float* xnext = xB;
    for (int l = 0; l < L_; ++l) {
        rmsnorm_kernel<<<B_, 256, 0, stream>>>(
            xcur, norm_w + (size_t)l * D_, xn, D_);

        // xz = xn @ in_proj_w[l].T   (O = 4096, K = 1024) — async-LDS staged
        gemm_wmma_staged<<<(2 * ED_) / 32, 32, 0, stream>>>(
            xn, D_, in_proj_w + (size_t)l * 2 * ED_ * D_, xz, 2 * ED_, D_);

        conv_kernel<<<(B_ * ED_) / 256, 256, 0, stream>>>(
            xz,
            conv_inputs + (size_t)l * B_ * ED_ * (K_ - 1),
            conv_w + (size_t)l * ED_ * K_,
            conv_b + (size_t)l * ED_,
            xc,
            out + CI_BASE + (size_t)l * B_ * ED_ * (K_ - 1));

        // dbc = xc @ x_proj_w[l].T   (O = 96, K = 2048)
        gemm_wmma_direct<0><<<(R_ + 2 * N_ + 15) / 16, 32, 0, stream>>>(
            xc, ED_, x_proj_w + (size_t)l * (R_ + 2 * N_) * ED_,
            nullptr, nullptr, dbc, R_ + 2 * N_, ED_);

        // delta = softplus(dbc[:, :R] @ dt_proj_w[l].T + dt_proj_b[l])
        gemm_wmma_direct<1><<<ED_ / 16, 32, 0, stream>>>(
            dbc, R_ + 2 * N_, dt_proj_w + (size_t)l * ED_ * R_,
            dt_proj_b + (size_t)l * ED_, nullptr, delta, ED_, R_);

        ssm_kernel<<<(B_ * ED_) / 256, 256, 0, stream>>>(
            delta, xc, xz, dbc,
            A_log + (size_t)l * ED_ * N_,
            D_param + (size_t)l * ED_,
            hs + (size_t)l * B_ * ED_ * N_,
            out + HS_BASE + (size_t)l * B_ * ED_ * N_,
            u);

        // x_new = u @ out_proj_w[l].T + x   (O = 1024, K = 2048)
        gemm_wmma_direct<2><<<D_ / 16, 32, 0, stream>>>(
            u, ED_, out_proj_w + (size_t)l * D_ * ED_,
            nullptr, xcur, xnext, D_, ED_);

        float* t = xcur; xcur = xnext; xnext = t;
    }

    // logits = rmsnorm(x, norm_f_w) @ embedding.T  (O = 50280, K = 1024) — staged
    rmsnorm_kernel<<<B_, 256, 0, stream>>>(xcur, norm_f_w, xn, D_);
    gemm_wmma_staged<<<(V_ + 31) / 32, 32, 0, stream>>>(
        xn, D_, embedding, out, V_, D_);
}